// CluserAssignment_7017976562073
// MI455X (gfx1250) — compile-verified
//
#include <hip/hip_runtime.h>
#include <hip/hip_bf16.h>

// Problem constants (match reference)
#define N_ROWS   16384
#define NUM_C    64      // clusters
#define NUM_K    256     // latent dim
#define SPLITK   16
#define KCHUNK   (N_ROWS / SPLITK)   // 1024 rows of K per block
#define TILE_N   16
#define KP       260     // padded LDS row stride: 16B-aligned rows, 260%64==4
#define CHALF    32      // clusters staged in LDS at a time

typedef __attribute__((ext_vector_type(2))) float v2f;
typedef __attribute__((ext_vector_type(8))) float v8f;

// ---------------------------------------------------------------------------
// Zero workspace accumulators (counts[64] followed by S[64*256])
// ---------------------------------------------------------------------------
__global__ void init_zero(float* __restrict__ p, int n) {
    int i = blockIdx.x * blockDim.x + threadIdx.x;
    if (i < n) p[i] = 0.0f;
}

// ---------------------------------------------------------------------------
// counts[c] = sum_n mask[n][c]   (mask is one-hot, mostly zeros)
// ---------------------------------------------------------------------------
__global__ __launch_bounds__(256) void counts_kernel(
    const float* __restrict__ mask, float* __restrict__ counts) {
    __shared__ float lc[NUM_C];
    const int t = threadIdx.x;
    if (t < NUM_C) lc[t] = 0.0f;
    __syncthreads();
    const size_t total  = (size_t)N_ROWS * NUM_C;
    const size_t stride = (size_t)gridDim.x * blockDim.x;
    for (size_t i = (size_t)blockIdx.x * blockDim.x + t; i < total; i += stride) {
        float v = mask[i];
        if (v != 0.0f) atomicAdd(&lc[(int)(i & (NUM_C - 1))], v);
    }
    __syncthreads();
    if (t < NUM_C) atomicAdd(&counts[t], lc[t]);
}

// ---------------------------------------------------------------------------
// S(64x256) += mask^T(64xN) @ X(Nx256) via V_WMMA_F32_16X16X4_F32.
// grid = (64 tiles, SPLITK), block = 32 threads (one wave, EXEC all ones).
// A-matrix 16x4 f32 layout (ISA 7.12.2): lanes 0-15 hold M=lane, VGPR0/1 = K0/K1;
// lanes 16-31 hold K2/K3. B mirrors A with N=lane&15. C/D: VGPR r -> row r (lanes
// 0-15) / row 8+r (lanes 16-31), col = lane&15.
// ---------------------------------------------------------------------------
__global__ __launch_bounds__(32) void centroid_gemm_wmma(
    const float* __restrict__ mask,   // N x 64
    const float* __restrict__ X,      // N x 256
    float* __restrict__ S) {          // 64 x 256 accumulators
    const int tile  = blockIdx.x;            // 0..63
    const int cBase = (tile >> 4) * 16;      // cluster tile (4 tiles)
    const int lBase = (tile & 15) * 16;      // latent tile  (16 tiles)
    const int lane  = threadIdx.x;
    const int m     = lane & 15;
    const int half  = lane >> 4;             // 0 -> K{0,1}, 1 -> K{2,3}
    const int kBeg  = blockIdx.y * KCHUNK;
    const int kEnd  = kBeg + KCHUNK;

    v8f acc = {};
    #pragma unroll 4
    for (int kk = kBeg; kk < kEnd; kk += 4) {
        const int k0 = kk + 2 * half;
        v2f a, b;
        a.x = mask[(size_t)(k0    ) * NUM_C + cBase + m];
        a.y = mask[(size_t)(k0 + 1) * NUM_C + cBase + m];
        b.x = X[(size_t)(k0    ) * NUM_K + lBase + m];
        b.y = X[(size_t)(k0 + 1) * NUM_K + lBase + m];
        acc = __builtin_amdgcn_wmma_f32_16x16x4_f32(
            /*neg_a=*/false, a, /*neg_b=*/false, b,
            /*c_mod=*/(short)0, acc, /*reuse_a=*/false, /*reuse_b=*/false);
    }
    #pragma unroll
    for (int r = 0; r < 8; ++r) {
        const int row = half * 8 + r;
        atomicAdd(&S[(size_t)(cBase + row) * NUM_K + lBase + m], acc[r]);
    }
}

// ---------------------------------------------------------------------------
// For each row n: dist[c] = sum_k |x[k] - mu[c][k]|, q = 1/(1+dist),
// out = q / sum_c(q).  mu staged 32 clusters at a time in padded LDS.
// All inner-loop LDS traffic is ds_load_b128 (rows 16B aligned); each thread
// owns one cluster and two X rows, so each mu float4 is reused twice.
// LDS: 32*260*4 + 16*256*4 + 16*64*4 + 64 = ~52.6 KB (< 64 KB static).
// ---------------------------------------------------------------------------
__global__ __launch_bounds__(256) void dist_norm_kernel(
    const float* __restrict__ X,       // N x 256
    const float* __restrict__ S,       // 64 x 256 centroid sums
    const float* __restrict__ counts,  // 64
    float* __restrict__ out) {         // N x 64
    __shared__ __align__(16) float mu[CHALF * KP];
    __shared__ __align__(16) float xs[TILE_N * NUM_K];
    __shared__ float qrow[TILE_N * NUM_C];
    __shared__ float rinv[TILE_N];

    const int t  = threadIdx.x;
    const int n0 = blockIdx.x * TILE_N;

    // Stage the 16-row X tile with b128 loads/stores.
    {
        const float4* Xg = reinterpret_cast<const float4*>(X + (size_t)n0 * NUM_K);
        float4*       xv = reinterpret_cast<float4*>(xs);
        for (int i = t; i < TILE_N * NUM_K / 4; i += 256) xv[i] = Xg[i];
    }

    const int c  = t & (CHALF - 1);      // lane -> cluster column
    const int rs = t >> 5;               // 0..7 -> rows rs and rs+8

    for (int h = 0; h < 2; ++h) {
        const int cB = h * CHALF;
        __syncthreads();  // xs ready (h=0) / previous mu consumers done (h=1)
        // Stage mu half: b128 from S, per-element divide by count, b128 to LDS.
        for (int i = t; i < CHALF * NUM_K / 4; i += 256) {
            const int cc = i >> 6;              // / (NUM_K/4)
            const int k4 = i & (NUM_K / 4 - 1);
            const float cnt = counts[cB + cc];
            float4 sv = reinterpret_cast<const float4*>(
                            S + (size_t)(cB + cc) * NUM_K)[k4];
            sv.x /= cnt; sv.y /= cnt; sv.z /= cnt; sv.w /= cnt;
            reinterpret_cast<float4*>(&mu[cc * KP])[k4] = sv;
        }
        __syncthreads();

        const float4* mr4 = reinterpret_cast<const float4*>(&mu[c * KP]);
        const float4* x04 = reinterpret_cast<const float4*>(&xs[(rs    ) * NUM_K]);
        const float4* x14 = reinterpret_cast<const float4*>(&xs[(rs + 8) * NUM_K]);
        float d0 = 0.0f, d1 = 0.0f;
        #pragma unroll 4
        for (int k4 = 0; k4 < NUM_K / 4; ++k4) {
            const float4 mv = mr4[k4];
            const float4 a  = x04[k4];
            const float4 b  = x14[k4];
            d0 += fabsf(a.x - mv.x); d1 += fabsf(b.x - mv.x);
            d0 += fabsf(a.y - mv.y); d1 += fabsf(b.y - mv.y);
            d0 += fabsf(a.z - mv.z); d1 += fabsf(b.z - mv.z);
            d0 += fabsf(a.w - mv.w); d1 += fabsf(b.w - mv.w);
        }
        qrow[(rs    ) * NUM_C + cB + c] = 1.0f / (1.0f + d0);
        qrow[(rs + 8) * NUM_C + cB + c] = 1.0f / (1.0f + d1);
    }
    __syncthreads();
    if (t < TILE_N) {
        float s = 0.0f;
        #pragma unroll
        for (int c2 = 0; c2 < NUM_C; ++c2) s += qrow[t * NUM_C + c2];
        rinv[t] = 1.0f / s;
    }
    __syncthreads();
    for (int i = t; i < TILE_N * NUM_C; i += 256)
        out[(size_t)n0 * NUM_C + i] = qrow[i] * rinv[i >> 6];
}

// ---------------------------------------------------------------------------
extern "C" void kernel_launch(void* const* d_in, const int* in_sizes, int n_in,
                              void* d_out, int out_size, void* d_ws, size_t ws_size,
                              hipStream_t stream) {
    const float* X    = (const float*)d_in[0];  // (N, 256)
    const float* mask = (const float*)d_in[1];  // (N, 64)
    float* out    = (float*)d_out;              // (N, 64)
    float* counts = (float*)d_ws;               // 64 floats
    float* S      = counts + NUM_C;             // 64*256 floats

    const int nz = NUM_C + NUM_C * NUM_K;
    init_zero<<<(nz + 255) / 256, 256, 0, stream>>>(counts, nz);
    counts_kernel<<<64, 256, 0, stream>>>(mask, counts);
    dim3 ggrid(64, SPLITK);
    centroid_gemm_wmma<<<ggrid, 32, 0, stream>>>(mask, X, S);
    dist_norm_kernel<<<N_ROWS / TILE_N, 256, 0, stream>>>(X, S, counts, out);
}